// BoundaryDeformation_32100585570630
// MI455X (gfx1250) — compile-verified
//
#include <hip/hip_runtime.h>
#include <hip/hip_bf16.h>

// ---------------------------------------------------------------------------
// BoundaryDeformation (1-D deformable attention) for MI455X / gfx1250
//   D=256, H=8, P2=8, DH=32, L=2048, B=16, NQ=300
// Memory-bound overall (~150MB traffic vs 9.4 GFLOP fp32) -> keep fp32 and
// run all four matmuls on V_WMMA_F32_16X16X4_F32.  Each wave owns a
// 16(M) x 64(N) tile = 4 independent WMMA accumulator chains, with
// software-pipelined loads (next K-slice in flight during current WMMAs).
// ---------------------------------------------------------------------------

typedef __attribute__((ext_vector_type(2))) float v2f;
typedef __attribute__((ext_vector_type(8))) float v8f;

#define CD_D   256
#define CD_H   8
#define CD_P2  8
#define CD_DH  32
#define CD_L   2048
#define CD_B   16
#define CD_NQ  300
#define CD_K1  512   // 2*D, K of value projection

__device__ __forceinline__ v8f wmma4(v2f a, v2f b, v8f c) {
  // (neg_a, A, neg_b, B, c_mod, C, reuse_a, reuse_b)
  return __builtin_amdgcn_wmma_f32_16x16x4_f32(false, a, false, b, (short)0, c,
                                               false, false);
}

// 16(M) x 64(N) x K tile.  Per ISA 7.12.2 layouts:
//   A 16x4 f32: lane m = lane&15, vgpr j holds K = 2*(lane>>4)+j
//   B 4x16 f32: lane n = lane&15, vgpr j holds K = 2*(lane>>4)+j
// Arow/Bj point at the per-lane row (A: row m, B: column-source row n).
// Rotating registers: loads for step i+1 issued before WMMAs of step i.
template <int K>
__device__ __forceinline__ void gemm_16x64(
    const float* __restrict__ Arow,
    const float* __restrict__ B0, const float* __restrict__ B1,
    const float* __restrict__ B2, const float* __restrict__ B3,
    int half, v8f acc[4])
{
  const int ka = 2 * half;
  __builtin_prefetch(Arow + (K / 2) + ka, 0, 3);   // global_prefetch_b8, no guard
  v2f a  = *(const v2f*)(Arow + ka);
  v2f b0 = *(const v2f*)(B0 + ka);
  v2f b1 = *(const v2f*)(B1 + ka);
  v2f b2 = *(const v2f*)(B2 + ka);
  v2f b3 = *(const v2f*)(B3 + ka);
#pragma unroll 4
  for (int k0 = 0; k0 < K - 4; k0 += 4) {
    const int kn = ka + k0 + 4;
    v2f an  = *(const v2f*)(Arow + kn);
    v2f bn0 = *(const v2f*)(B0 + kn);
    v2f bn1 = *(const v2f*)(B1 + kn);
    v2f bn2 = *(const v2f*)(B2 + kn);
    v2f bn3 = *(const v2f*)(B3 + kn);
    acc[0] = wmma4(a, b0, acc[0]);
    acc[1] = wmma4(a, b1, acc[1]);
    acc[2] = wmma4(a, b2, acc[2]);
    acc[3] = wmma4(a, b3, acc[3]);
    a = an; b0 = bn0; b1 = bn1; b2 = bn2; b3 = bn3;
  }
  acc[0] = wmma4(a, b0, acc[0]);
  acc[1] = wmma4(a, b1, acc[1]);
  acc[2] = wmma4(a, b2, acc[2]);
  acc[3] = wmma4(a, b3, acc[3]);
}

// ---------------------------------------------------------------------------
// Kernel A: value projection
//   features [L*B, 512] @ W_val^T + b_val -> V [B,H,L,32]
// waves = 2048 M-tiles * 4 N-groups = 8192
// ---------------------------------------------------------------------------
__global__ __launch_bounds__(128) void bd_value_proj(
    const float* __restrict__ features,
    const float* __restrict__ W_val,      // [256, 512]
    const float* __restrict__ b_val,
    float* __restrict__ V)                // [16,8,2048,32]
{
  const int wave = blockIdx.x * (blockDim.x >> 5) + (threadIdx.x >> 5);
  const int lane = threadIdx.x & 31;
  const int ng   = wave & 3;             // N-group: n = ng*64 + j*16 + l15
  const int mt   = wave >> 2;            // 0..2047
  const int half = lane >> 4;
  const int l15  = lane & 15;

  const float* __restrict__ Arow = features + (size_t)(mt * 16 + l15) * CD_K1;
  const float* __restrict__ B0 = W_val + (size_t)(ng * 64 +  0 + l15) * CD_K1;
  const float* __restrict__ B1 = W_val + (size_t)(ng * 64 + 16 + l15) * CD_K1;
  const float* __restrict__ B2 = W_val + (size_t)(ng * 64 + 32 + l15) * CD_K1;
  const float* __restrict__ B3 = W_val + (size_t)(ng * 64 + 48 + l15) * CD_K1;

  v8f acc[4] = {};
  gemm_16x64<CD_K1>(Arow, B0, B1, B2, B3, half, acc);

#pragma unroll
  for (int j = 0; j < 4; ++j) {
    const int d    = ng * 64 + j * 16 + l15;
    const float bi = b_val[d];
    const int h    = d >> 5;
    const int dh   = d & 31;
#pragma unroll
    for (int i = 0; i < 8; ++i) {
      const int r  = mt * 16 + i + 8 * half;   // r = l*16 + b
      const int l  = r >> 4;
      const int bb = r & 15;
      V[(((size_t)bb * CD_H + h) * CD_L + l) * CD_DH + dh] = acc[j][i] + bi;
    }
  }
}

// ---------------------------------------------------------------------------
// Kernel B: query projection (offsets ++ attn logits) as one GEMM
//   query rows r = b*300+nq (A row = pro_features[nq][b][:]),
//   B rows: n<64 -> W_off[n], n>=64 -> W_attn[n-64]
//   -> raw [4800, 128]  (cols 0..63 offsets+bias, 64..127 logits+bias)
// waves = 300 M-tiles * 2 N-groups = 600
// ---------------------------------------------------------------------------
__global__ __launch_bounds__(128) void bd_qproj(
    const float* __restrict__ pro_features,  // [300,16,256]
    const float* __restrict__ W_off,  const float* __restrict__ b_off,
    const float* __restrict__ W_attn, const float* __restrict__ b_attn,
    float* __restrict__ raw)                 // [4800,128]
{
  const int wave = blockIdx.x * (blockDim.x >> 5) + (threadIdx.x >> 5);
  const int lane = threadIdx.x & 31;
  const int ng   = wave & 1;
  const int mt   = wave >> 1;              // 0..299
  const int half = lane >> 4;
  const int l15  = lane & 15;

  // A row: r = mt*16 + l15 -> (b, nq) = (r/300, r%300); row data at [nq][b]
  const int r0   = mt * 16 + l15;
  const int b0_  = r0 / CD_NQ;
  const int nq0  = r0 % CD_NQ;
  const float* __restrict__ Arow =
      pro_features + ((size_t)nq0 * CD_B + b0_) * CD_D;

  const float* Bp[4];
  float bias[4];
#pragma unroll
  for (int j = 0; j < 4; ++j) {
    const int n = ng * 64 + j * 16 + l15;
    if (n < 64) { Bp[j] = W_off  + (size_t)n * CD_D;        bias[j] = b_off[n]; }
    else        { Bp[j] = W_attn + (size_t)(n - 64) * CD_D; bias[j] = b_attn[n - 64]; }
  }

  v8f acc[4] = {};
  gemm_16x64<CD_D>(Arow, Bp[0], Bp[1], Bp[2], Bp[3], half, acc);

#pragma unroll
  for (int j = 0; j < 4; ++j) {
    const int n = ng * 64 + j * 16 + l15;
#pragma unroll
    for (int i = 0; i < 8; ++i) {
      const int r = mt * 16 + i + 8 * half;   // r = b*300+nq, store direct
      raw[(size_t)r * 128 + n] = acc[j][i] + bias[j];
    }
  }
}

// ---------------------------------------------------------------------------
// Kernel C: per-head softmax, location math, bilinear 1-D gather from V,
// weighted sum -> mid [B,NQ,256].  One block (256 threads) per (b,nq);
// 8 heads x 32 dh-lanes -> each gather is a contiguous 128B per wave.
// ---------------------------------------------------------------------------
__global__ __launch_bounds__(256) void bd_sample(
    const float* __restrict__ raw,       // [4800,128]
    const float* __restrict__ boundary,  // [16,300,2]
    const float* __restrict__ V,         // [16,8,2048,32]
    float* __restrict__ mid)             // [B,NQ,256]
{
  __shared__ float off_s[64];
  __shared__ float logit_s[64];
  __shared__ float ix_s[64];
  __shared__ float w_s[64];

  const int bid = blockIdx.x;            // b*NQ + nq
  const int b   = bid / CD_NQ;
  const int nq  = bid % CD_NQ;
  const int tid = threadIdx.x;

  if (tid < 64)        off_s[tid]        = raw[(size_t)bid * 128 + tid];
  else if (tid < 128)  logit_s[tid - 64] = raw[(size_t)bid * 128 + tid];
  __syncthreads();

  if (tid < CD_H) {
    const int h = tid;
    const float center = boundary[(size_t)bid * 2 + 0];
    const float width  = boundary[(size_t)bid * 2 + 1];
    const float start  = center - 0.5f * width;
    const float end    = center + 0.5f * width;

    float mx = -3.4e38f;
#pragma unroll
    for (int p = 0; p < CD_P2; ++p) mx = fmaxf(mx, logit_s[h * CD_P2 + p]);
    float sum = 0.f, e[CD_P2];
#pragma unroll
    for (int p = 0; p < CD_P2; ++p) {
      e[p] = __expf(logit_s[h * CD_P2 + p] - mx);
      sum += e[p];
    }
    const float inv = 1.0f / sum;
#pragma unroll
    for (int p = 0; p < CD_P2; ++p) {
      const float base = (p & 1) ? end : start;          // [s,e,s,e,...]
      float loc = base + off_s[h * CD_P2 + p] * (width * 0.5f);
      loc = fminf(fmaxf(loc, 0.0f), 1.0f);
      // grid = 2*loc-1 ; ix = (grid+1)*L/2 - 0.5 = loc*L - 0.5
      ix_s[h * CD_P2 + p] = loc * (float)CD_L - 0.5f;
      w_s[h * CD_P2 + p]  = e[p] * inv;
    }
  }
  __syncthreads();

  const int h  = tid >> 5;
  const int dh = tid & 31;
  const float* __restrict__ Vh = V + ((size_t)b * CD_H + h) * (size_t)CD_L * CD_DH;
  float acc = 0.f;
#pragma unroll
  for (int p = 0; p < CD_P2; ++p) {
    const float ix  = ix_s[h * CD_P2 + p];
    const float wt  = w_s[h * CD_P2 + p];
    const float x0f = floorf(ix);
    const float fr  = ix - x0f;
    const int x0 = (int)x0f;
    const int x1 = x0 + 1;
    const float g0 = (x0 >= 0 && x0 < CD_L) ? Vh[(size_t)x0 * CD_DH + dh] : 0.f;
    const float g1 = (x1 >= 0 && x1 < CD_L) ? Vh[(size_t)x1 * CD_DH + dh] : 0.f;
    acc = fmaf(wt, fmaf(g1 - g0, fr, g0), acc);   // wt*(g0*(1-fr)+g1*fr)
  }
  mid[(size_t)bid * CD_D + tid] = acc;
}

// ---------------------------------------------------------------------------
// Kernel D: output projection  mid [4800,256] @ W_out^T + b_out
// row remap (b,nq) -> out[nq][b][:].  waves = 300 * 4 = 1200
// ---------------------------------------------------------------------------
__global__ __launch_bounds__(128) void bd_out_proj(
    const float* __restrict__ mid,     // [4800, 256] row r = b*300+nq
    const float* __restrict__ W_out,   // [256, 256]
    const float* __restrict__ b_out,
    float* __restrict__ out)           // [NQ,B,256] row = nq*16+b
{
  const int wave = blockIdx.x * (blockDim.x >> 5) + (threadIdx.x >> 5);
  const int lane = threadIdx.x & 31;
  const int ng   = wave & 3;
  const int mt   = wave >> 2;          // 0..299
  const int half = lane >> 4;
  const int l15  = lane & 15;

  const float* __restrict__ Arow = mid + (size_t)(mt * 16 + l15) * CD_D;
  const float* __restrict__ B0 = W_out + (size_t)(ng * 64 +  0 + l15) * CD_D;
  const float* __restrict__ B1 = W_out + (size_t)(ng * 64 + 16 + l15) * CD_D;
  const float* __restrict__ B2 = W_out + (size_t)(ng * 64 + 32 + l15) * CD_D;
  const float* __restrict__ B3 = W_out + (size_t)(ng * 64 + 48 + l15) * CD_D;

  v8f acc[4] = {};
  gemm_16x64<CD_D>(Arow, B0, B1, B2, B3, half, acc);

#pragma unroll
  for (int j = 0; j < 4; ++j) {
    const int d    = ng * 64 + j * 16 + l15;
    const float bi = b_out[d];
#pragma unroll
    for (int i = 0; i < 8; ++i) {
      const int r  = mt * 16 + i + 8 * half;   // r = b*300 + nq
      const int bb = r / CD_NQ;
      const int q  = r % CD_NQ;
      out[((size_t)q * CD_B + bb) * CD_D + d] = acc[j][i] + bi;
    }
  }
}

// ---------------------------------------------------------------------------
// Launch
// ---------------------------------------------------------------------------
extern "C" void kernel_launch(void* const* d_in, const int* in_sizes, int n_in,
                              void* d_out, int out_size, void* d_ws, size_t ws_size,
                              hipStream_t stream) {
  const float* pro_features = (const float*)d_in[0];   // [300,16,256]
  const float* features     = (const float*)d_in[1];   // [2048,16,512]
  const float* boundary     = (const float*)d_in[2];   // [16,300,2]
  const float* W_off        = (const float*)d_in[3];   // [64,256]
  const float* b_off        = (const float*)d_in[4];   // [64]
  const float* W_attn       = (const float*)d_in[5];   // [64,256]
  const float* b_attn       = (const float*)d_in[6];   // [64]
  const float* W_val        = (const float*)d_in[7];   // [256,512]
  const float* b_val        = (const float*)d_in[8];   // [256]
  const float* W_out        = (const float*)d_in[9];   // [256,256]
  const float* b_out        = (const float*)d_in[10];  // [256]
  (void)in_sizes; (void)n_in; (void)out_size; (void)ws_size;

  float* V   = (float*)d_ws;                                   // 8,388,608 f
  float* mid = V + (size_t)CD_B * CD_H * CD_L * CD_DH;         // 1,228,800 f
  float* raw = mid + (size_t)CD_B * CD_NQ * CD_D;              //   614,400 f

  // A: 8192 waves / 4 per block
  bd_value_proj<<<2048, 128, 0, stream>>>(features, W_val, b_val, V);
  // B: 600 waves / 4 per block
  bd_qproj<<<150, 128, 0, stream>>>(pro_features, W_off, b_off, W_attn, b_attn, raw);
  // C: one block per (b,nq)
  bd_sample<<<CD_B * CD_NQ, 256, 0, stream>>>(raw, boundary, V, mid);
  // D: 1200 waves / 4 per block
  bd_out_proj<<<300, 128, 0, stream>>>(mid, W_out, b_out, (float*)d_out);
}